// Quantization_27771258536790
// MI455X (gfx1250) — compile-verified
//
#include <hip/hip_runtime.h>

// ---------------------------------------------------------------------------
// Dequantize: out[g*4 + j] = codebook[(codes[g] + 256*(g>=HALF))*4 + j] * scales[g/16]
// Memory-bound: ~324 MB total traffic -> ~14 us at 23.3 TB/s. No matmul
// structure, so the optimal MI455X path is b128 streaming + LDS-resident LUT.
// ---------------------------------------------------------------------------

typedef __attribute__((ext_vector_type(4))) float f32x4;
typedef __attribute__((ext_vector_type(4))) int   i32x4;

// Address-space-qualified vector pointee types for the async-LDS builtin
// (clang prints AS1 as "__device__" — builtin wants typed int4 pointers).
typedef __attribute__((address_space(1))) i32x4 gi32x4;   // global
typedef __attribute__((address_space(3))) i32x4 li32x4;   // LDS

#define CODEBOOK_NUM  2
#define CENTROIDS_NUM 256
#define CENTROID_LEN  4
#define ROWS          8192
#define COLUMNS       8192

constexpr long long NUMEL     = (long long)ROWS * COLUMNS;          // 67,108,864
constexpr unsigned  NGROUPS   = (unsigned)(NUMEL / CENTROID_LEN);   // 16,777,216
constexpr unsigned  HALF_GRPS = NGROUPS / CODEBOOK_NUM;             //  8,388,608
constexpr int       LUT_VECS  = CODEBOOK_NUM * CENTROIDS_NUM;       // 512 float4 = 8 KB
constexpr int       TPB       = 256;                                // 8 waves (wave32)
constexpr int       ILP       = 4;
constexpr int       NBLOCKS   = (int)(NGROUPS / (TPB * ILP));       // 16384

static_assert((long long)NBLOCKS * TPB * ILP == (long long)NGROUPS, "exact tiling");

#if __has_builtin(__builtin_amdgcn_global_load_async_to_lds_b128)
#define USE_ASYNC_LDS 1
#else
#define USE_ASYNC_LDS 0
#endif

__global__ __launch_bounds__(TPB) void Quantization_27771258536790_kernel(
    const f32x4* __restrict__ cb,      // [512]  flattened codebooks (float4 centroids)
    const float* __restrict__ scales,  // [NUMEL/64]
    const int*   __restrict__ codes,   // [NGROUPS] (codebook rows concatenated)
    f32x4*       __restrict__ out)     // [NGROUPS]
{
    __shared__ f32x4 lut[LUT_VECS];    // 8 KB, resident per workgroup

    // ---- Stage codebook into LDS -----------------------------------------
#if USE_ASYNC_LDS
    // gfx1250 async DMA path: global -> LDS without VGPR round-trip.
    for (int i = threadIdx.x; i < LUT_VECS; i += TPB) {
        gi32x4* gsrc = (gi32x4*)(unsigned long long)(cb + i);
        li32x4* ldst = (li32x4*)(unsigned)(unsigned long long)(&lut[i]);
        __builtin_amdgcn_global_load_async_to_lds_b128(gsrc, ldst, 0, 0);
    }
#if __has_builtin(__builtin_amdgcn_s_wait_asynccnt)
    __builtin_amdgcn_s_wait_asynccnt(0);
#else
    asm volatile("s_wait_asynccnt 0" ::: "memory");
#endif
#else
    for (int i = threadIdx.x; i < LUT_VECS; i += TPB)
        lut[i] = cb[i];                 // global_load_b128 + ds_store_b128
#endif
    __syncthreads();

    // ---- Streaming dequant loop ------------------------------------------
    // Grid-stride layout: per wave, codes reads are 128B-coalesced and
    // output stores are contiguous 512B b128 bursts.
    const unsigned base   = blockIdx.x * TPB + threadIdx.x;
    const unsigned stride = (unsigned)NBLOCKS * TPB;

#pragma unroll
    for (int k = 0; k < ILP; ++k) {
        const unsigned g = base + (unsigned)k * stride;

        // codes are streamed exactly once -> non-temporal
        const int code = __builtin_nontemporal_load(codes + g);
        const int idx  = code + ((g >= HALF_GRPS) ? CENTROIDS_NUM : 0);

        f32x4 v = lut[idx];             // ds_load_b128 gather (8 KB table)

        // 16 consecutive lanes share one scale -> broadcast-friendly load
        const float s = scales[g >> 4];
        v *= s;

        // 256 MB single-touch output: NT store to avoid rinsing L2
        __builtin_nontemporal_store(v, out + g);
    }
}

extern "C" void kernel_launch(void* const* d_in, const int* in_sizes, int n_in,
                              void* d_out, int out_size, void* d_ws, size_t ws_size,
                              hipStream_t stream) {
    (void)in_sizes; (void)n_in; (void)out_size; (void)d_ws; (void)ws_size;
    const f32x4* cb     = (const f32x4*)d_in[0];   // codebooks  [2,256,4] f32
    const float* scales = (const float*)d_in[1];   // scales     [numel/64] f32
    const int*   codes  = (const int*)d_in[2];     // codes      [2, numel/8] i32
    f32x4*       out    = (f32x4*)d_out;           // [8192*8192] f32 as float4

    Quantization_27771258536790_kernel<<<NBLOCKS, TPB, 0, stream>>>(cb, scales, codes, out);
}